// Conv_lstm_52166672777876
// MI455X (gfx1250) — compile-verified
//
#include <hip/hip_runtime.h>

typedef __attribute__((ext_vector_type(2))) float v2f;
typedef __attribute__((ext_vector_type(8))) float v8f;

#define B_SZ   512
#define T_IN   1125
#define E_IN   5
#define XROW   (T_IN * E_IN)      // 5625 floats per batch
#define KT     25                 // conv_time kernel
#define POOLK  25
#define LSEQ   216                // pooled sequence length
#define NCH    40
#define KDIM   245                // 49 taps * 5 electrodes
#define KPAD   248                // K padded to multiple of 4
#define NPAD   48                 // N padded to multiple of 16
#define HID    10
#define G4     40                 // 4*HID
#define LTILES 14                 // ceil(216/16)
#define XS_PAD 5824               // max LDS x index = 25*223+247 = 5822

// workspace layout (in floats)
#define WS_W2P_OFF 0                    // KPAD*NPAD interleaved-pair weights
#define WS_C2_OFF  (KPAD * NPAD)        // 40 floats
#define WS_XG_OFF  (KPAD * NPAD + 64)   // xg[LSEQ][B_SZ][G4]

// ---------------------------------------------------------------------------
// gfx1250 async global->LDS copy (ASYNCcnt path), with safe fallback.
// Probe told us: param0 is addrspace(1) int* ("__device__ int *"), 4 args.
// ---------------------------------------------------------------------------
#if defined(__has_builtin)
#if __has_builtin(__builtin_amdgcn_global_load_async_to_lds_b32)
#define HAVE_ASYNC_LDS 1
#endif
#endif

typedef __attribute__((address_space(1))) int as1_int;
typedef __attribute__((address_space(3))) int as3_int;

__device__ __forceinline__ void g2l_b32(const float* __restrict__ g,
                                        float* __restrict__ l_) {
#ifdef HAVE_ASYNC_LDS
  __builtin_amdgcn_global_load_async_to_lds_b32(
      (as1_int*)g, (as3_int*)l_, 0, 0);
#else
  *l_ = *g;
#endif
}
__device__ __forceinline__ void g2l_wait() {
#ifdef HAVE_ASYNC_LDS
#if __has_builtin(__builtin_amdgcn_s_wait_asynccnt)
  __builtin_amdgcn_s_wait_asynccnt(0);
#else
  asm volatile("s_wait_asynccnt 0x0" ::: "memory");
#endif
#endif
}

// ---------------------------------------------------------------------------
// Kernel 1: fold conv_time + conv_spat + BN + avgpool + w_ih/biases into
//           W2p[KPAD][NPAD] (interleaved K-pairs for b64 LDS loads) and c2[40].
// ---------------------------------------------------------------------------
__global__ void prep_kernel(const float* __restrict__ tw,   // [40][25]
                            const float* __restrict__ tb,   // [40]
                            const float* __restrict__ sw,   // [40][40][5]
                            const float* __restrict__ gamma,
                            const float* __restrict__ beta,
                            const float* __restrict__ mean,
                            const float* __restrict__ var,
                            const float* __restrict__ wih,  // [40][40]
                            const float* __restrict__ bih,
                            const float* __restrict__ bhh,
                            float* __restrict__ ws) {
  __shared__ float weff[NCH * KT * E_IN];   // 5000
  __shared__ float wp[NCH * KDIM];          // 9800
  __shared__ float cf[NCH];
  __shared__ float scl[NCH];
  const int tid = threadIdx.x;

  // Weff[f,k,e] = sum_ci spat[f,ci,e] * time[ci,k]
  for (int idx = tid; idx < NCH * KT * E_IN; idx += blockDim.x) {
    int f = idx / (KT * E_IN);
    int r = idx % (KT * E_IN);
    int k = r / E_IN, e = r % E_IN;
    float s = 0.f;
    for (int ci = 0; ci < NCH; ++ci)
      s += sw[(f * NCH + ci) * E_IN + e] * tw[ci * KT + k];
    weff[idx] = s;
  }
  // per-channel BN scale and pooled constant
  if (tid < NCH) {
    float s = 0.f;
    for (int ci = 0; ci < NCH; ++ci) {
      float tbv = tb[ci];
      for (int e = 0; e < E_IN; ++e)
        s += sw[(tid * NCH + ci) * E_IN + e] * tbv;
    }
    float sc = gamma[tid] * rsqrtf(var[tid] + 1e-5f);
    scl[tid] = sc;
    cf[tid]  = s * sc + (beta[tid] - mean[tid] * sc);
  }
  __syncthreads();

  // Wp[f,j,e] = scale_f/25 * sum over pool window of Weff
  for (int idx = tid; idx < NCH * KDIM; idx += blockDim.x) {
    int f = idx / KDIM;
    int r = idx % KDIM;
    int j = r / E_IN, e = r % E_IN;
    int k0 = j - (POOLK - 1); if (k0 < 0) k0 = 0;
    int k1 = j;               if (k1 > KT - 1) k1 = KT - 1;
    float s = 0.f;
    for (int k = k0; k <= k1; ++k) s += weff[(f * KT + k) * E_IN + e];
    wp[idx] = s * scl[f] * (1.f / 25.f);
  }
  __syncthreads();

  // W2p[kk][g] = sum_f wih[g,f] * Wp[f,kk]; store interleaved K-pairs
  for (int idx = tid; idx < KPAD * NPAD; idx += blockDim.x) {
    int kk = idx / NPAD;
    int g  = idx % NPAD;
    float v = 0.f;
    if (kk < KDIM && g < NCH) {
      float s = 0.f;
      for (int f = 0; f < NCH; ++f) s += wih[g * NCH + f] * wp[f * KDIM + kk];
      v = s;
    }
    ws[WS_W2P_OFF + ((kk >> 1) * NPAD + g) * 2 + (kk & 1)] = v;
  }
  // c2[g] = b_ih + b_hh + sum_f wih[g,f] * cf[f]
  if (tid < NCH) {
    float s = bih[tid] + bhh[tid];
    for (int f = 0; f < NCH; ++f) s += wih[tid * NCH + f] * cf[f];
    ws[WS_C2_OFF + tid] = s;
  }
}

// ---------------------------------------------------------------------------
// Kernel 2: fused conv front-end as fp32 WMMA GEMM.
// One workgroup = one batch (448 threads = 14 waves = 14 l-tiles).
// x row (22.5KB) + weights (47.6KB) staged in LDS; inner loop is pure
// ds_load + v_wmma_f32_16x16x4_f32.
// xg[l][b][g] = X[l,:245] . W2[:,g] + c2[g],  X row l = x[b, 25l : 25l+245]
// ---------------------------------------------------------------------------
__device__ __forceinline__ void store_tile(float* __restrict__ xg, v8f acc,
                                           int lt, int half, int g, float bias,
                                           int b) {
  union { v8f v; float f[8]; } u; u.v = acc;
#pragma unroll
  for (int r = 0; r < 8; ++r) {
    int lr = lt * 16 + r + 8 * half;
    if (lr < LSEQ)
      xg[((size_t)lr * B_SZ + b) * G4 + g] = u.f[r] + bias;
  }
}

__global__ __launch_bounds__(448) void conv_wmma_kernel(
    const float* __restrict__ x, const float* __restrict__ ws,
    float* __restrict__ xg) {
  __shared__ float xs[XS_PAD];         // one batch row of x, zero-padded tail
  __shared__ float bsm[KPAD * NPAD];   // interleaved K-pair weight layout
  __shared__ float c2s[NCH];
  const int tid = threadIdx.x;
  const int b   = blockIdx.x;

  const float* __restrict__ xb = x + (size_t)b * XROW;
  for (int i = tid; i < XROW; i += 448)          g2l_b32(xb + i, &xs[i]);
  for (int i = XROW + tid; i < XS_PAD; i += 448) xs[i] = 0.f;
  for (int i = tid; i < KPAD * NPAD; i += 448)   g2l_b32(ws + WS_W2P_OFF + i,
                                                         &bsm[i]);
  if (tid < NCH) c2s[tid] = ws[WS_C2_OFF + tid];
  g2l_wait();
  __syncthreads();

  const int lt   = tid >> 5;           // wave id == l-tile 0..13
  const int lane = tid & 31;
  const int half = lane >> 4;          // K sub-group: 0 -> K0/K1, 1 -> K2/K3
  const int mn   = lane & 15;          // A row (M) and B col (N) index
  const int arow = 25 * (lt * 16 + mn);

  v8f acc0 = {}, acc1 = {}, acc2 = {};
#pragma unroll 4
  for (int k0 = 0; k0 < KPAD; k0 += 4) {
    const int ka = k0 + 2 * half;
    v2f a;
    a.x = xs[arow + ka];
    a.y = xs[arow + ka + 1];
    const float* __restrict__ bp = &bsm[((ka >> 1) * NPAD + mn) * 2];
    v2f b0, b1, b2;
    b0.x = bp[0];  b0.y = bp[1];       // N tile 0: cols  0..15
    b1.x = bp[32]; b1.y = bp[33];      // N tile 1: cols 16..31
    b2.x = bp[64]; b2.y = bp[65];      // N tile 2: cols 32..47
    acc0 = __builtin_amdgcn_wmma_f32_16x16x4_f32(false, a, false, b0,
                                                 (short)0, acc0, false, false);
    acc1 = __builtin_amdgcn_wmma_f32_16x16x4_f32(false, a, false, b1,
                                                 (short)0, acc1, false, false);
    acc2 = __builtin_amdgcn_wmma_f32_16x16x4_f32(false, a, false, b2,
                                                 (short)0, acc2, false, false);
  }

  // D layout: VGPR r -> row M = r + 8*half, col N = mn
  { int g = mn;      if (g < NCH) store_tile(xg, acc0, lt, half, g, c2s[g], b); }
  { int g = 16 + mn; if (g < NCH) store_tile(xg, acc1, lt, half, g, c2s[g], b); }
  { int g = 32 + mn; if (g < NCH) store_tile(xg, acc2, lt, half, g, c2s[g], b); }
}

// ---------------------------------------------------------------------------
// Kernel 3: LSTM recurrence + FC head. One wave32 per batch element,
// all cross-gate traffic via __shfl (ds_bpermute) -- zero barriers.
// Lane ln owns gate ln; lanes 0..7 also own gate 32+ln. Lanes 0..9 own c[j].
// ---------------------------------------------------------------------------
__device__ __forceinline__ float sigmoidf_(float x) {
  return 1.f / (1.f + __expf(-x));
}
__device__ __forceinline__ float tanhf_(float x) {
  x = fminf(15.f, fmaxf(-15.f, x));
  float e = __expf(2.f * x);
  return (e - 1.f) / (e + 1.f);
}

__global__ __launch_bounds__(256) void lstm_kernel(
    const float* __restrict__ xg, const float* __restrict__ whh,  // [40][10]
    const float* __restrict__ fcw, const float* __restrict__ fcb, // [2][10],[2]
    float* __restrict__ out) {
  const int lane = threadIdx.x & 31;
  const int b    = blockIdx.x * 8 + (threadIdx.x >> 5);

  const int g1row = (lane < 8) ? (32 + lane) : 32;   // clamped (unused lanes)
  float w0[HID], w1[HID], h[HID];
#pragma unroll
  for (int j = 0; j < HID; ++j) {
    w0[j] = whh[lane * HID + j];
    w1[j] = whh[g1row * HID + j];
    h[j]  = 0.f;
  }
  float c = 0.f;
  const bool is_g0 = (lane >= 2 * HID) && (lane < 3 * HID);  // tanh gate rows

  for (int l = 0; l < LSEQ; ++l) {
    const float* __restrict__ xr = xg + ((size_t)l * B_SZ + b) * G4;
    float p0 = xr[lane];
    float p1 = (lane < 8) ? xr[32 + lane] : 0.f;
#pragma unroll
    for (int j = 0; j < HID; ++j) {
      p0 += h[j] * w0[j];
      p1 += h[j] * w1[j];
    }
    float a0 = is_g0 ? tanhf_(p0) : sigmoidf_(p0);
    float a1 = sigmoidf_(p1);                  // gates 32..39 are all 'o'
    // owner lane j<10: i=a0(j), f=a0(j+10), g=a0(j+20), o=gate 30+j
    float f_ = __shfl(a0, lane + 10, 32);
    float g_ = __shfl(a0, lane + 20, 32);
    float oa = __shfl(a0, lane + 30, 32);      // gates 30,31 (lanes 0,1)
    float ob = __shfl(a1, (lane >= 2) ? (lane - 2) : 0, 32);  // gates 32..39
    float o_ = (lane < 2) ? oa : ob;
    c = f_ * c + a0 * g_;
    float hn = o_ * tanhf_(c);
#pragma unroll
    for (int j = 0; j < HID; ++j) h[j] = __shfl(hn, j, 32);
  }

  if (lane < 2) {
    float s = fcb[lane];
#pragma unroll
    for (int j = 0; j < HID; ++j) s += h[j] * fcw[lane * HID + j];
    out[b * 2 + lane] = s;
  }
}

// ---------------------------------------------------------------------------
extern "C" void kernel_launch(void* const* d_in, const int* in_sizes, int n_in,
                              void* d_out, int out_size, void* d_ws,
                              size_t ws_size, hipStream_t stream) {
  (void)in_sizes; (void)n_in; (void)out_size; (void)ws_size;
  const float* x   = (const float*)d_in[0];
  const float* tw  = (const float*)d_in[1];
  const float* tb  = (const float*)d_in[2];
  const float* sw  = (const float*)d_in[3];
  const float* gam = (const float*)d_in[4];
  const float* bet = (const float*)d_in[5];
  const float* mea = (const float*)d_in[6];
  const float* var = (const float*)d_in[7];
  const float* wih = (const float*)d_in[8];
  const float* whh = (const float*)d_in[9];
  const float* bih = (const float*)d_in[10];
  const float* bhh = (const float*)d_in[11];
  const float* fcw = (const float*)d_in[12];
  const float* fcb = (const float*)d_in[13];

  float* ws = (float*)d_ws;
  float* xg = ws + WS_XG_OFF;   // [216][512][40] floats

  prep_kernel<<<1, 256, 0, stream>>>(tw, tb, sw, gam, bet, mea, var,
                                     wih, bih, bhh, ws);
  conv_wmma_kernel<<<B_SZ, 448, 0, stream>>>(x, ws, xg);
  lstm_kernel<<<B_SZ / 8, 256, 0, stream>>>(xg, whh, fcw, fcb, (float*)d_out);
}